// OSCwSTU_72576357368007
// MI455X (gfx1250) — compile-verified
//
#include <hip/hip_runtime.h>
#include <hip/hip_bf16.h>
#include <cstddef>

// Problem dimensions (fixed by the reference)
constexpr int T_  = 512;
constexpr int N_  = 1024;
constexpr int MC_ = 512;
constexpr int H_  = 5;
constexpr int M_  = 20;
constexpr int NF_ = 20;
constexpr int J_  = NF_ * MC_;   // 10240 : concatenated (f,c) dimension

typedef __attribute__((ext_vector_type(2))) float v2f;
typedef __attribute__((ext_vector_type(8))) float v8f;

// ---------------------------------------------------------------------------
// Generic fp32 WMMA GEMM:  C[M,N] = A[M,K] * op(B)  (+ optional per-col bias)
// op(B) = B (row-major K x N) if TRANSB==0, else B^T with B stored N x K.
// One wave computes one 16x16 tile of C with V_WMMA_F32_16X16X4_F32;
// k-loop unrolled x4 (K=16/trip, all K dims are multiples of 16).
// A-layout (ISA 7.12.2, 32-bit A 16x4): lane%16 = M row, lane/16 selects K pair.
// C/D-layout: VGPR r -> M = r (lanes 0-15) / r+8 (lanes 16-31), N = lane%16.
// TRANSB is a template parameter so the inner loop is branch-free.
// ---------------------------------------------------------------------------
template <int TRANSB>
__global__ void gemm_wmma_f32(const float* __restrict__ A,
                              const float* __restrict__ B,
                              float* __restrict__ C,
                              const float* __restrict__ bias,
                              int Kdim, int lda, int ldb, int ldc)
{
    const int tileN = blockIdx.x * 16;
    const int tileM = blockIdx.y * 16;
    const int lane  = threadIdx.x & 31;
    const int half  = lane >> 4;       // 0 or 1
    const int l16   = lane & 15;
    const int rowA  = tileM + l16;
    const int colB  = tileN + l16;
    const int koff  = half * 2;

    const float* arow = A + (size_t)rowA * lda + koff;

    v8f acc = {};
    for (int kb = 0; kb < Kdim; kb += 16) {
        v2f a[4], b[4];
        #pragma unroll
        for (int u = 0; u < 4; ++u) {
            const float* ap = arow + kb + 4 * u;
            a[u].x = ap[0];
            a[u].y = ap[1];
            if (TRANSB) {
                const float* bp = B + (size_t)colB * ldb + kb + 4 * u + koff;
                b[u].x = bp[0];
                b[u].y = bp[1];
            } else {
                b[u].x = B[(size_t)(kb + 4 * u + koff)     * ldb + colB];
                b[u].y = B[(size_t)(kb + 4 * u + koff + 1) * ldb + colB];
            }
        }
        #pragma unroll
        for (int u = 0; u < 4; ++u)
            acc = __builtin_amdgcn_wmma_f32_16x16x4_f32(
                      false, a[u], false, b[u], (short)0, acc, false, false);
    }

    if (bias) {
        float bv = bias[colB];
        #pragma unroll
        for (int r = 0; r < 8; ++r)
            C[(size_t)(tileM + r + half * 8) * ldc + colB] = acc[r] + bv;
    } else {
        #pragma unroll
        for (int r = 0; r < 8; ++r)
            C[(size_t)(tileM + r + half * 8) * ldc + colB] = acc[r];
    }
}

// ---------------------------------------------------------------------------
// Xw[t, i*N+n] = sum_k phi[k,i] * W[t+k-(M-1), n]   (zero-padded on the left)
// ---------------------------------------------------------------------------
__global__ void build_Xw(const float* __restrict__ W,
                         const float* __restrict__ phi,
                         float* __restrict__ Xw)
{
    int idx = blockIdx.x * blockDim.x + threadIdx.x;   // over T*H*N
    if (idx >= T_ * H_ * N_) return;
    int n = idx % N_;
    int i = (idx / N_) % H_;
    int t = idx / (N_ * H_);
    float s = 0.f;
    #pragma unroll
    for (int k = 0; k < M_; ++k) {
        int src = t + k - (M_ - 1);
        if (src >= 0) s += phi[k * H_ + i] * W[(size_t)src * N_ + n];
    }
    Xw[idx] = s;
}

// Et[(i*N+n), c] = sigma[i]^0.25 * E[c,n,i]      (E is (MC,N,H), i innermost)
__global__ void build_Et(const float* __restrict__ E,
                         const float* __restrict__ sigma,
                         float* __restrict__ Et)
{
    int idx = blockIdx.x * blockDim.x + threadIdx.x;   // over H*N*MC
    if (idx >= H_ * N_ * MC_) return;
    int c = idx % MC_;
    int n = (idx / MC_) % N_;
    int i = idx / (MC_ * N_);
    float sc = sqrtf(sqrtf(sigma[i]));                 // sigma > 0 (top Hankel eigs)
    Et[idx] = sc * E[((size_t)c * N_ + n) * H_ + i];
}

__global__ void zero_f32(float* __restrict__ p, int n)
{
    int i = blockIdx.x * blockDim.x + threadIdx.x;
    if (i < n) p[i] = 0.f;
}

// ---------------------------------------------------------------------------
// Sequential step, part 1:  u_t[c2] = u_pert[t,c2] - Pcat[c2,:] . Acat
// One block per output row; Pcat (21 MB) stays resident in the 192 MB L2.
// ---------------------------------------------------------------------------
__global__ void step_matvec(const float* __restrict__ Pcat,
                            const float* __restrict__ Acat,
                            const float* __restrict__ u_pert,
                            float* __restrict__ Uall, int t)
{
    __shared__ float red[256];
    const int c2 = blockIdx.x;
    const float* prow = Pcat + (size_t)c2 * J_;
    float s = 0.f;
    for (int j = threadIdx.x; j < J_; j += 256)
        s += prow[j] * Acat[j];
    red[threadIdx.x] = s;
    __syncthreads();
    for (int off = 128; off > 0; off >>= 1) {
        if (threadIdx.x < off) red[threadIdx.x] += red[threadIdx.x + off];
        __syncthreads();
    }
    if (threadIdx.x == 0)
        Uall[(size_t)t * MC_ + c2] = u_pert[(size_t)t * MC_ + c2] - red[0];
}

// Sequential step, part 2: snapshot Acat(t) then rank-1 update with u_t.
__global__ void step_update(float* __restrict__ Acat,
                            float* __restrict__ Ahist,
                            const float* __restrict__ Uall,
                            const float* __restrict__ phi_stu, int t)
{
    int j = blockIdx.x * blockDim.x + threadIdx.x;
    if (j >= J_) return;
    float a = Acat[j];
    Ahist[(size_t)t * J_ + j] = a;               // = Acat before step-t update
    int f = j / MC_, c = j % MC_;
    Acat[j] = a + phi_stu[t * NF_ + f] * Uall[(size_t)t * MC_ + c];
}

// losses[t] = X[t,:].Y[t,:] + U[t,:].Z[t,:]
__global__ void loss_kernel(const float* __restrict__ X, const float* __restrict__ Y,
                            const float* __restrict__ U, const float* __restrict__ Z,
                            float* __restrict__ out)
{
    __shared__ float red[256];
    const int t = blockIdx.x;
    float s = 0.f;
    for (int n = threadIdx.x; n < N_; n += 256)
        s += X[(size_t)t * N_ + n] * Y[(size_t)t * N_ + n];
    for (int c = threadIdx.x; c < MC_; c += 256)
        s += U[(size_t)t * MC_ + c] * Z[(size_t)t * MC_ + c];
    red[threadIdx.x] = s;
    __syncthreads();
    for (int off = 128; off > 0; off >>= 1) {
        if (threadIdx.x < off) red[threadIdx.x] += red[threadIdx.x + off];
        __syncthreads();
    }
    if (threadIdx.x == 0) out[t] = red[0];
}

// ---------------------------------------------------------------------------
extern "C" void kernel_launch(void* const* d_in, const int* in_sizes, int n_in,
                              void* d_out, int out_size, void* d_ws, size_t ws_size,
                              hipStream_t stream)
{
    // inputs: Q, R, K, E, bias, E_stu, phi, sigma, phi_stu, W
    const float* Q       = (const float*)d_in[0];
    const float* R       = (const float*)d_in[1];
    const float* K       = (const float*)d_in[2];
    const float* E       = (const float*)d_in[3];
    const float* bias    = (const float*)d_in[4];
    const float* E_stu   = (const float*)d_in[5];   // (NF, MC, N) == (J_, N_) 2-D view
    const float* phi     = (const float*)d_in[6];
    const float* sigma   = (const float*)d_in[7];
    const float* phi_stu = (const float*)d_in[8];
    const float* W       = (const float*)d_in[9];
    float* out = (float*)d_out;

    // workspace layout (floats)
    float* ws     = (float*)d_ws;
    float* Xw     = ws;                              // 512*5120
    float* Et     = Xw     + (size_t)T_ * H_ * N_;   // 5120*512
    float* u_pert = Et     + (size_t)H_ * N_ * MC_;  // 512*512
    float* Pcat   = u_pert + (size_t)T_ * MC_;       // 512*10240
    float* Acat   = Pcat   + (size_t)MC_ * J_;       // 10240
    float* Ahist  = Acat   + (size_t)J_;             // 512*10240
    float* Uall   = Ahist  + (size_t)T_ * J_;        // 512*512
    float* Xb     = Uall   + (size_t)T_ * MC_;       // 512*1024
    float* Yb     = Xb     + (size_t)T_ * N_;        // 512*1024
    float* Zb     = Yb     + (size_t)T_ * N_;        // 512*512

    // 0) init recurrence state
    zero_f32<<<(J_ + 255) / 256, 256, 0, stream>>>(Acat, J_);

    // 1) gather kernels for the u_pert GEMM
    build_Xw<<<(T_ * H_ * N_ + 255) / 256, 256, 0, stream>>>(W, phi, Xw);
    build_Et<<<(H_ * N_ * MC_ + 255) / 256, 256, 0, stream>>>(E, sigma, Et);

    // 2) u_pert(512x512) = Xw(512x5120) @ Et(5120x512) + bias   [WMMA]
    gemm_wmma_f32<0><<<dim3(MC_ / 16, T_ / 16), 32, 0, stream>>>(
        Xw, Et, u_pert, bias, H_ * N_, H_ * N_, MC_, MC_);

    // 3) Pcat(512x10240) = K(512x1024) @ E_stu^T (E_stu is J_ x N_)   [WMMA]
    gemm_wmma_f32<1><<<dim3(J_ / 16, MC_ / 16), 32, 0, stream>>>(
        K, E_stu, Pcat, nullptr, N_, N_, N_, J_);

    // 4) sequential feedback loop: only 5.2M MAC/step, Pcat L2-resident
    for (int t = 0; t < T_; ++t) {
        step_matvec<<<MC_, 256, 0, stream>>>(Pcat, Acat, u_pert, Uall, t);
        step_update<<<(J_ + 255) / 256, 256, 0, stream>>>(Acat, Ahist, Uall, phi_stu, t);
    }

    // 5) X(512x1024) = Ahist(512x10240) @ E_stu(10240x1024)   [WMMA]
    gemm_wmma_f32<0><<<dim3(N_ / 16, T_ / 16), 32, 0, stream>>>(
        Ahist, E_stu, Xb, nullptr, J_, J_, N_, N_);

    // 6) Y = X @ Q ; Z = U @ R   [WMMA]
    gemm_wmma_f32<0><<<dim3(N_ / 16, T_ / 16), 32, 0, stream>>>(
        Xb, Q, Yb, nullptr, N_, N_, N_, N_);
    gemm_wmma_f32<0><<<dim3(MC_ / 16, T_ / 16), 32, 0, stream>>>(
        Uall, R, Zb, nullptr, MC_, MC_, MC_, MC_);

    // 7) losses[t] = x.Qx + u.Ru
    loss_kernel<<<T_, 256, 0, stream>>>(Xb, Yb, Uall, Zb, out);
}